// QNetwork_66709432042071
// MI455X (gfx1250) — compile-verified
//
#include <hip/hip_runtime.h>
#include <hip/hip_bf16.h>
#include <math.h>

#define NQ      12
#define NL      5
#define NPAIRS  6        // fused 2-qubit gates per layer
#define NSTATE  4096     // 2^12
#define THREADS 256      // 8 waves of 32
#define ODIM    4
#define BATCH   1024

typedef __attribute__((ext_vector_type(2))) float v2f;
typedef __attribute__((ext_vector_type(8))) float v8f;

// Inverse of the CNOT-ring basis permutation in flat-bit space
// (flat bit = 11 - qubit index; reference puts qubit 0 at the MSB).
__device__ __forceinline__ unsigned cnot_ring_inv(unsigned v) {
    v ^= (v & 1u) << 11;            // b11 ^= b0
#pragma unroll
    for (int j = 0; j < 11; ++j)    // b[j] ^= b[j+1]
        v ^= ((v >> (j + 1)) & 1u) << j;
    return v;
}

// Complex 2x2 single-qubit unitary U = M(theta) * Rx(iw * tanh(x))
__device__ __forceinline__ void make_u(const float* __restrict__ x,
                                       const float* __restrict__ iw,
                                       const float* __restrict__ theta,
                                       int b, int l, int q,
                                       float Ur[2][2], float Ui[2][2]) {
    const float xt = tanhf(x[b * NQ + q]);
    const float a  = iw[l * NQ + q] * xt;
    const float bb = theta[(l * NQ + q) * 2 + 0];
    const float cc = theta[(l * NQ + q) * 2 + 1];
    float sa, ca, sb, cb, sc, cs;
    sincosf(0.5f * a,  &sa, &ca);
    sincosf(0.5f * bb, &sb, &cb);
    sincosf(0.5f * cc, &sc, &cs);
    const float ecr = cs, eci = -sc;                 // exp(-i c/2)
    const float M00r =  ecr * cb, M00i =  eci * cb;
    const float M01r = -ecr * sb, M01i = -eci * sb;
    const float M10r =  ecr * sb, M10i = -eci * sb;
    const float M11r =  ecr * cb, M11i = -eci * cb;
    // U = M @ Rx,  Rx = [[ca, -i sa], [-i sa, ca]];  z*(-i sa) = (sa*zi, -sa*zr)
    Ur[0][0] = M00r * ca + sa * M01i;  Ui[0][0] = M00i * ca - sa * M01r;
    Ur[0][1] = sa * M00i + M01r * ca;  Ui[0][1] = -sa * M00r + M01i * ca;
    Ur[1][0] = M10r * ca + sa * M11i;  Ui[1][0] = M10i * ca - sa * M11r;
    Ur[1][1] = sa * M10i + M11r * ca;  Ui[1][1] = -sa * M10r + M11i * ca;
}

__global__ __launch_bounds__(THREADS)
void qsim_kernel(const float* __restrict__ x,      // [1024,12]
                 const float* __restrict__ iw,     // [5,12]
                 const float* __restrict__ theta,  // [5,12,2]
                 const float* __restrict__ ow,     // [4]
                 float* __restrict__ out)          // [1024,4]
{
    __shared__ float2 st[NSTATE];                 // 32 KB statevector
    __shared__ float  G2[NL * NPAIRS][64];        // 7.5 KB: 30 fused 8x8 real gates
    __shared__ float  red[ODIM * THREADS];        // 4 KB reduction scratch

    const int b = blockIdx.x;
    const int t = threadIdx.x;

    // ---- init state to |0...0> ----
    for (int i = t; i < NSTATE; i += THREADS) st[i] = make_float2(0.f, 0.f);
    if (t == 0) st[0] = make_float2(1.f, 0.f);

    // ---- build 30 fused two-qubit gates: G2 = embed8x8( U_{2j} (x) U_{2j+1} ) ----
    if (t < NL * NPAIRS) {
        const int l = t / NPAIRS, j = t % NPAIRS;
        float U0r[2][2], U0i[2][2], U1r[2][2], U1i[2][2];
        make_u(x, iw, theta, b, l, 2 * j,     U0r, U0i);  // acts on high bit p_hi
        make_u(x, iw, theta, b, l, 2 * j + 1, U1r, U1i);  // acts on low bit p_lo
        float* g = G2[t];
#pragma unroll
        for (int s = 0; s < 4; ++s) {
            const int a = s >> 1, bq = s & 1;
#pragma unroll
            for (int sp = 0; sp < 4; ++sp) {
                const int ap = sp >> 1, bp = sp & 1;
                const float kr = U0r[a][ap] * U1r[bq][bp] - U0i[a][ap] * U1i[bq][bp];
                const float ki = U0r[a][ap] * U1i[bq][bp] + U0i[a][ap] * U1r[bq][bp];
                g[(2 * s    ) * 8 + 2 * sp    ] =  kr;
                g[(2 * s    ) * 8 + 2 * sp + 1] = -ki;
                g[(2 * s + 1) * 8 + 2 * sp    ] =  ki;
                g[(2 * s + 1) * 8 + 2 * sp + 1] =  kr;
            }
        }
    }
    __syncthreads();

    const int  lane = t & 31;
    const int  wave = t >> 5;
    const int  nlo  = lane & 15;       // WMMA column = quad index within group of 16
    const bool hi   = lane >= 16;      // low lanes: B K=0,1; high lanes: B K=2,3
    const int  koff = hi ? 2 : 0;
    const int  row  = lane & 7;        // G2 row for the A fragment

    for (int l = 0; l < NL; ++l) {
        // ---- 6 fused two-qubit gates, each = 2 chained V_WMMA_F32_16X16X4_F32 ----
        for (int jg = 0; jg < NPAIRS; ++jg) {
            const int p_lo = 10 - 2 * jg;                // adjacent flat bits (p_lo, p_lo+1)
            const float* g = G2[l * NPAIRS + jg];
            v2f a1, a2;                                  // A = 2 stacked copies of 8x4 slices
            a1.x = g[row * 8 + koff];      a1.y = g[row * 8 + koff + 1];
            a2.x = g[row * 8 + 4 + koff];  a2.y = g[row * 8 + 4 + koff + 1];
            const int ld1 = hi ? (1 << p_lo) : 0;              // i01 : i00  -> B1
            const int ld2 = ld1 | (2 << p_lo);                 // i11 : i10  -> B2
            const int st1 = hi ? (2 << p_lo) : 0;              // i10 : i00

            // 8 waves x 8 chains x 16 quads = 1024 quads = whole state
            int qd   = (wave * 8) * 16 + nlo;
            int base = ((qd >> p_lo) << (p_lo + 2)) | (qd & ((1 << p_lo) - 1));
            float2 b1 = st[base + ld1];
            float2 b2 = st[base + ld2];
            for (int it = 0; it < 8; ++it) {
                int    nbase = base;
                float2 nb1 = b1, nb2 = b2;
                if (it < 7) {                       // software-pipelined prefetch
                    const int nqd = (wave * 8 + it + 1) * 16 + nlo;
                    nbase = ((nqd >> p_lo) << (p_lo + 2)) | (nqd & ((1 << p_lo) - 1));
                    nb1 = st[nbase + ld1];
                    nb2 = st[nbase + ld2];
                }
                v2f bf1; bf1.x = b1.x; bf1.y = b1.y;
                v2f bf2; bf2.x = b2.x; bf2.y = b2.y;
                v8f acc = {};
                acc = __builtin_amdgcn_wmma_f32_16x16x4_f32(
                    false, a1, false, bf1, (short)0, acc, false, false);
                acc = __builtin_amdgcn_wmma_f32_16x16x4_f32(
                    false, a2, false, bf2, (short)0, acc, false, false);
                float2 r1, r2;
                r1.x = hi ? acc[4] : acc[0];  r1.y = hi ? acc[5] : acc[1];
                r2.x = hi ? acc[6] : acc[2];  r2.y = hi ? acc[7] : acc[3];
                st[base + st1]                 = r1;   // i00 / i10
                st[base + st1 + (1 << p_lo)]   = r2;   // i01 / i11
                base = nbase; b1 = nb1; b2 = nb2;
            }
            __syncthreads();
        }
        // ---- fused CNOT ring: one gather through B^{-1} ----
        float2 tmp[16];
#pragma unroll
        for (int k = 0; k < 16; ++k) {
            const unsigned idx = (unsigned)(t + k * THREADS);
            tmp[k] = st[cnot_ring_inv(idx)];
        }
        __syncthreads();
#pragma unroll
        for (int k = 0; k < 16; ++k) st[t + k * THREADS] = tmp[k];
        __syncthreads();
    }

    // ---- Z expectations for qubits 0..3, then scale by output_weights ----
    float z[ODIM] = {0.f, 0.f, 0.f, 0.f};
    for (int k = 0; k < 16; ++k) {
        const int idx = t + k * THREADS;
        const float2 a2v = st[idx];
        const float pr2 = a2v.x * a2v.x + a2v.y * a2v.y;
#pragma unroll
        for (int j = 0; j < ODIM; ++j)
            z[j] += ((idx >> (11 - j)) & 1) ? -pr2 : pr2;
    }
#pragma unroll
    for (int j = 0; j < ODIM; ++j) red[j * THREADS + t] = z[j];
    __syncthreads();
    for (int s = THREADS / 2; s > 0; s >>= 1) {
        if (t < s) {
#pragma unroll
            for (int j = 0; j < ODIM; ++j)
                red[j * THREADS + t] += red[j * THREADS + t + s];
        }
        __syncthreads();
    }
    if (t < ODIM) out[b * ODIM + t] = ow[t] * red[t * THREADS];
}

extern "C" void kernel_launch(void* const* d_in, const int* in_sizes, int n_in,
                              void* d_out, int out_size, void* d_ws, size_t ws_size,
                              hipStream_t stream) {
    (void)in_sizes; (void)n_in; (void)out_size; (void)d_ws; (void)ws_size;
    const float* x     = (const float*)d_in[0];  // [1024,12]
    const float* iw    = (const float*)d_in[1];  // [5,12]
    const float* theta = (const float*)d_in[2];  // [5,12,2]
    const float* ow    = (const float*)d_in[3];  // [4]
    float* out = (float*)d_out;                  // [1024,4]
    qsim_kernel<<<BATCH, THREADS, 0, stream>>>(x, iw, theta, ow, out);
}